// constrained_network_28157805592680
// MI455X (gfx1250) — compile-verified
//
#include <hip/hip_runtime.h>
#include <hip/hip_bf16.h>
#include <math.h>

#define NN 20000
#define EE 640000
#define CC 64
#define EMBD 32
#define NBAS 8
#define NLAY 2
#define RHID 32

typedef float v2f __attribute__((ext_vector_type(2)));
typedef float v8f __attribute__((ext_vector_type(8)));
typedef unsigned int u32x4 __attribute__((ext_vector_type(4)));
typedef int i32x4 __attribute__((ext_vector_type(4)));
typedef int i32x8 __attribute__((ext_vector_type(8)));

// ---------------------------------------------------------------------------
// TDM 2-D tile load: Global -> LDS via tensor_load_to_lds (TENSORcnt).
// D# layout per CDNA5 ISA 8.3/8.4: group0 = {flags, lds_addr, global_addr,
// addr_hi|type}, group1 = {data_size, tensor dims, tile dims, strides}.
// tile_w contiguous elements per row (float32), tile_h rows, row stride in
// elements. Groups 2/3 zero (2-D tensor).
// ---------------------------------------------------------------------------
__device__ __forceinline__ void tdm_load_2d_f32(unsigned int lds_off, const float* gptr,
                                                unsigned int tensor_w, unsigned int tensor_h,
                                                unsigned long long row_stride,
                                                unsigned int tile_w, unsigned int tile_h) {
  unsigned long long ga = (unsigned long long)(uintptr_t)gptr;
  u32x4 g0;
  g0[0] = 1u;                                                  // count=1 (valid user D#)
  g0[1] = lds_off;                                             // lds_addr (bytes)
  g0[2] = (unsigned int)ga;                                    // global_addr[31:0]
  g0[3] = ((unsigned int)(ga >> 32) & 0x01FFFFFFu) | (2u << 30);  // addr[56:32] | type=2
  i32x8 g1;
  g1[0] = (int)(2u << 16);                                     // data_size = 4 bytes
  g1[1] = (int)((tensor_w & 0xFFFFu) << 16);                   // tensor_dim0[15:0]
  g1[2] = (int)(((tensor_w >> 16) & 0xFFFFu) | ((tensor_h & 0xFFFFu) << 16));
  g1[3] = (int)(((tensor_h >> 16) & 0xFFFFu) | (tile_w << 16));  // tile_dim0
  g1[4] = (int)(tile_h & 0xFFFFu);                             // tile_dim1 (tile_dim2=0)
  g1[5] = (int)(unsigned int)(row_stride & 0xFFFFFFFFull);     // dim0_stride[31:0]
  g1[6] = (int)(unsigned int)((row_stride >> 32) & 0xFFFFull); // dim0_stride[47:32]
  g1[7] = 0;
  i32x4 gz;  gz[0] = gz[1] = gz[2] = gz[3] = 0;
#if __clang_major__ >= 23
  i32x8 gz8;
  for (int i = 0; i < 8; ++i) gz8[i] = 0;
  __builtin_amdgcn_tensor_load_to_lds(g0, g1, gz, gz, gz8, 0);
#else
  __builtin_amdgcn_tensor_load_to_lds(g0, g1, gz, gz, 0);
#endif
}

__device__ __forceinline__ unsigned int lds_off_of(const void* p) {
  // generic LDS pointer = {shared aperture, addr[31:0] = wave-relative offset}
  return (unsigned int)(uintptr_t)p;
}

// ---------------------------------------------------------------------------
// Q = polar factor of uplift_M (64x2): Q = M * (M^T M)^{-1/2}
// ---------------------------------------------------------------------------
__global__ void compute_Q(const float* __restrict__ M, float* __restrict__ Q) {
  __shared__ float R[4];
  if (threadIdx.x == 0) {
    float a = 0.f, b = 0.f, d = 0.f;
    for (int c = 0; c < CC; ++c) {
      float m0 = M[2 * c], m1 = M[2 * c + 1];
      a += m0 * m0; b += m0 * m1; d += m1 * m1;
    }
    float det = a * d - b * b;
    float s = sqrtf(fmaxf(det, 0.f));
    float t = sqrtf(fmaxf(a + d + 2.f * s, 1e-30f));
    float a2 = (a + s) / t, b2 = b / t, d2 = (d + s) / t;
    float inv = 1.f / (a2 * d2 - b2 * b2);
    R[0] = d2 * inv; R[1] = -b2 * inv; R[2] = -b2 * inv; R[3] = a2 * inv;
  }
  __syncthreads();
  int c = threadIdx.x;
  if (c < CC) {
    float m0 = M[2 * c], m1 = M[2 * c + 1];
    Q[2 * c]     = m0 * R[0] + m1 * R[2];
    Q[2 * c + 1] = m0 * R[1] + m1 * R[3];
  }
}

// ---------------------------------------------------------------------------
// init: emb gather, y_s = 0, y_v = einsum('nkx,ck->ncx', xv, Q), y_old = y
// ---------------------------------------------------------------------------
__global__ void init_state(const float* __restrict__ x, const int* __restrict__ node_attr,
                           const float* __restrict__ emb_table, const float* __restrict__ Q,
                           float* __restrict__ emb,
                           float* __restrict__ ys, float* __restrict__ yolds,
                           float* __restrict__ yv, float* __restrict__ yoldv) {
  int t = blockIdx.x * blockDim.x + threadIdx.x;
  int n = t >> 6, c = t & 63;
  float q0 = Q[2 * c], q1 = Q[2 * c + 1];
  const float* xr = x + n * 6;
#pragma unroll
  for (int xd = 0; xd < 3; ++xd) {
    float v = q0 * xr[xd] + q1 * xr[3 + xd];
    yv[n * 192 + c * 3 + xd] = v;
    yoldv[n * 192 + c * 3 + xd] = v;
  }
  ys[t] = 0.f;
  yolds[t] = 0.f;
  if (c < EMBD) emb[n * EMBD + c] = emb_table[node_attr[n] * EMBD + c];
}

// ---------------------------------------------------------------------------
// Generic WMMA f32 GEMM: C[16x16 tile] (+)= A(MxK) @ B(KxNcols), K <= 64.
// One wave per tile. B tile (Kx16) DMA'd to LDS by the Tensor Data Mover,
// synchronized with s_wait_tensorcnt; K stepped by 4 via v_wmma_f32_16x16x4_f32.
// ---------------------------------------------------------------------------
__global__ __launch_bounds__(32) void wmma_gemm(
    const float* __restrict__ A, int lda, int aes,
    const float* __restrict__ B, int ldb,
    float* __restrict__ C, int ldc, int ces,
    int K, int accum) {
  __shared__ float sB[64 * 16];
  int lane = threadIdx.x & 31;
  int lo = lane & 15, hi = lane >> 4;
  long mbase = (long)blockIdx.x * 16;
  int nbase = blockIdx.y * 16;

  // DMA B[0:K, nbase:nbase+16] -> LDS (row-major, 16 contiguous floats/row)
  tdm_load_2d_f32(lds_off_of(sB), B + nbase, (unsigned)ldb, (unsigned)K,
                  (unsigned long long)ldb, 16u, (unsigned)K);

  const float* Arow = A + (mbase + lo) * (long)lda;
  v8f acc;
  if (accum) {
#pragma unroll
    for (int j = 0; j < 8; ++j)
      acc[j] = C[(mbase + j + 8 * hi) * (long)ldc + (long)(nbase + lo) * ces];
  } else {
#pragma unroll
    for (int j = 0; j < 8; ++j) acc[j] = 0.f;
  }

  __builtin_amdgcn_s_wait_tensorcnt(0);

  for (int k0 = 0; k0 < K; k0 += 4) {
    int k = k0 + 2 * hi;
    v2f a, b;
    a.x = Arow[(long)k * aes];
    a.y = Arow[(long)(k + 1) * aes];
    b.x = sB[k * 16 + lo];
    b.y = sB[(k + 1) * 16 + lo];
    acc = __builtin_amdgcn_wmma_f32_16x16x4_f32(false, a, false, b, (short)0, acc,
                                                false, false);
  }
#pragma unroll
  for (int j = 0; j < 8; ++j)
    C[(mbase + j + 8 * hi) * (long)ldc + (long)(nbase + lo) * ces] = acc[j];
}

// ---------------------------------------------------------------------------
// WMMA sc contraction: C += P @ W where P[n, c*32+e] = S[n,c] * Emb[n,e].
// S tile (16x64) and Emb tile (16x32) staged in LDS; W (2048 x Ncols) streamed
// as 32 chunks of 64x16 f32, double-buffered through LDS by the TDM
// (issue chunk c+1, s_wait_tensorcnt 1 -> chunk c resident, in-order).
// ---------------------------------------------------------------------------
__global__ __launch_bounds__(32) void wmma_sc_gemm(
    const float* __restrict__ S, int lds_, int ses,
    const float* __restrict__ Emb,
    const float* __restrict__ W, int ldw,
    float* __restrict__ C, int ldc, int ces) {
  __shared__ float sS[16][CC];
  __shared__ float sE[16][EMBD];
  __shared__ float sW[2][64 * 16];
  int lane = threadIdx.x & 31;
  long mbase = (long)blockIdx.x * 16;
  int nbase = blockIdx.y * 16;

  // prologue: DMA chunk 0 of W while we stage S/Emb tiles with vector loads
  tdm_load_2d_f32(lds_off_of(sW[0]), W + nbase, (unsigned)ldw, 2048u,
                  (unsigned long long)ldw, 16u, 64u);

  for (int idx = lane; idx < 16 * CC; idx += 32) {
    int r = idx >> 6, c = idx & 63;
    sS[r][c] = S[(mbase + r) * (long)lds_ + (long)c * ses];
  }
  for (int idx = lane; idx < 16 * EMBD; idx += 32) {
    int r = idx >> 5, c = idx & 31;
    sE[r][c] = Emb[(mbase + r) * EMBD + c];
  }
  __syncthreads();

  int lo = lane & 15, hi = lane >> 4;
  v8f acc;
#pragma unroll
  for (int j = 0; j < 8; ++j)
    acc[j] = C[(mbase + j + 8 * hi) * (long)ldc + (long)(nbase + lo) * ces];

  auto compute_chunk = [&](int ch, int buf) {
    const float* wb = sW[buf];
    int kb = ch * 64;
#pragma unroll 4
    for (int kk = 0; kk < 64; kk += 4) {
      int kr = kk + 2 * hi;          // row within chunk
      int k = kb + kr;               // global K index
      int k1 = k + 1;
      v2f a, b;
      a.x = sS[lo][k >> 5] * sE[lo][k & 31];
      a.y = sS[lo][k1 >> 5] * sE[lo][k1 & 31];
      b.x = wb[kr * 16 + lo];
      b.y = wb[(kr + 1) * 16 + lo];
      acc = __builtin_amdgcn_wmma_f32_16x16x4_f32(false, a, false, b, (short)0, acc,
                                                  false, false);
    }
  };

  for (int ch = 0; ch < 31; ++ch) {
    // issue next chunk into the other buffer, then wait for current (in-order)
    tdm_load_2d_f32(lds_off_of(sW[(ch + 1) & 1]), W + (long)(ch + 1) * 64 * ldw + nbase,
                    (unsigned)ldw, 2048u, (unsigned long long)ldw, 16u, 64u);
    __builtin_amdgcn_s_wait_tensorcnt(1);
    compute_chunk(ch, ch & 1);
  }
  __builtin_amdgcn_s_wait_tensorcnt(0);
  compute_chunk(31, 1);

#pragma unroll
  for (int j = 0; j < 8; ++j)
    C[(mbase + j + 8 * hi) * (long)ldc + (long)(nbase + lo) * ces] = acc[j];
}

// ---------------------------------------------------------------------------
// Edge kernel: one wave32 per edge (8 edges / 256-thread block; E = 8*80000).
// Geometry + cutoff + radial basis, 8->32->128 radial MLP (hidden via LDS),
// msg_s / msg_v scatter with global f32 atomics (pre-scaled by 1/sqrt(32)).
// ---------------------------------------------------------------------------
__global__ __launch_bounds__(256) void edge_message(
    const float* __restrict__ pos6,
    const int* __restrict__ esrc, const int* __restrict__ edst,
    const float* __restrict__ W1, const float* __restrict__ b1,
    const float* __restrict__ W2,
    const float* __restrict__ s1, const float* __restrict__ v1,
    float* __restrict__ agg_s, float* __restrict__ agg_v) {
  __shared__ float hid[8][RHID];
  const float PI = 3.14159265358979f;
  int wave = threadIdx.x >> 5, lane = threadIdx.x & 31;
  int e = blockIdx.x * 8 + wave;
  int src = esrc[e], dst = edst[e];

  float ev0 = pos6[src * 6 + 0] - pos6[dst * 6 + 0];
  float ev1 = pos6[src * 6 + 1] - pos6[dst * 6 + 1];
  float ev2 = pos6[src * 6 + 2] - pos6[dst * 6 + 2];
  float elen = sqrtf(ev0 * ev0 + ev1 * ev1 + ev2 * ev2);
  float safe = fmaxf(elen, 1e-9f);
  const float SQ3 = 1.7320508075688772f;
  float inv = 1.f / safe;
  float sh0 = SQ3 * ev0 * inv, sh1 = SQ3 * ev1 * inv, sh2 = SQ3 * ev2 * inv;
  float u = 2.f * (elen * 0.4f - 1.f);  // MAX_RADIUS = 2.5
  float cut = (u > 0.f) ? 0.f : ((u < -2.f) ? 1.f : 0.5f * (1.f - __cosf(PI * u)));
  float ea0 = cut * sh0, ea1 = cut * sh1, ea2 = cut * sh2;

  float ef[NBAS];
  float coef = 0.8944271909999159f * 2.8284271247461903f * inv;
  float arg = PI * safe * 0.4f;
#pragma unroll
  for (int k = 0; k < NBAS; ++k) ef[k] = coef * __sinf((float)(k + 1) * arg);

  float hj = b1[lane];
#pragma unroll
  for (int k = 0; k < NBAS; ++k) hj += ef[k] * W1[k * RHID + lane];
  hj = hj / (1.f + __expf(-hj));
  hid[wave][lane] = hj;
  __syncthreads();

  const float inv_nn = 0.17677669529663687f;  // 1/sqrt(32)
#pragma unroll
  for (int t = 0; t < 2; ++t) {
    int c = lane + t * 32;
    float w1c = 0.f, w2c = 0.f;
#pragma unroll 8
    for (int m = 0; m < RHID; ++m) {
      float h = hid[wave][m];
      w1c += h * W2[m * 128 + c];
      w2c += h * W2[m * 128 + CC + c];
    }
    float s1c = s1[src * CC + c];
    const float* vr = v1 + src * 192 + c * 3;
    float msg_s = w2c * (vr[0] * ea0 + vr[1] * ea1 + vr[2] * ea2) * inv_nn;
    atomicAdd(&agg_s[dst * CC + c], msg_s);
    float tv = w1c * s1c * inv_nn;
    atomicAdd(&agg_v[dst * 192 + c * 3 + 0], tv * ea0);
    atomicAdd(&agg_v[dst * 192 + c * 3 + 1], tv * ea1);
    atomicAdd(&agg_v[dst * 192 + c * 3 + 2], tv * ea2);
  }
}

// ---------------------------------------------------------------------------
// finalize: gates + residual leapfrog update; writes new state into the
// y_old buffers (host rotates pointers). One thread per (node, channel).
// ---------------------------------------------------------------------------
__global__ void finalize(const float* __restrict__ out_s, const float* __restrict__ out_v,
                         const float* __restrict__ y2s, const float* __restrict__ y2v,
                         const float* __restrict__ ys, const float* __restrict__ yv,
                         float* __restrict__ yolds, float* __restrict__ yoldv,
                         const float* __restrict__ hArr, const float* __restrict__ mixArr,
                         int layer) {
  int t = blockIdx.x * blockDim.x + threadIdx.x;
  int n = t >> 6, c = t & 63;
  float hv = hArr[layer];
  float hh = hv * hv;
  float m = mixArr[layer];
  float os = out_s[n * 128 + c];
  float gs = os / (1.f + __expf(-os));
  float og = out_s[n * 128 + CC + c];
  float gate = 1.f / (1.f + __expf(-og));
  float news = 2.f * ys[t] - yolds[t] + hh * (m * gs + (m - 1.f) * y2s[t]);
  yolds[t] = news;
#pragma unroll
  for (int xd = 0; xd < 3; ++xd) {
    int iv = n * 192 + c * 3 + xd;
    float gv = gate * out_v[iv];
    float newv = 2.f * yv[iv] - yoldv[iv] + hh * (m * gv + (m - 1.f) * y2v[iv]);
    yoldv[iv] = newv;
  }
}

// ---------------------------------------------------------------------------
// x_out[n, k*3+x] = sum_c new_v[n,c,x] * Q[c,k]
// ---------------------------------------------------------------------------
__global__ void project_x(const float* __restrict__ newv, const float* __restrict__ Q,
                          float* __restrict__ xout) {
  int t = blockIdx.x * blockDim.x + threadIdx.x;
  if (t >= NN * 6) return;
  int n = t / 6, r = t % 6;
  int k = r / 3, xd = r % 3;
  float s = 0.f;
#pragma unroll 8
  for (int c = 0; c < CC; ++c) s += newv[n * 192 + c * 3 + xd] * Q[c * 2 + k];
  xout[t] = s;
}

// ---------------------------------------------------------------------------
extern "C" void kernel_launch(void* const* d_in, const int* in_sizes, int n_in,
                              void* d_out, int out_size, void* d_ws, size_t ws_size,
                              hipStream_t stream) {
  (void)in_sizes; (void)n_in; (void)out_size; (void)ws_size;
  const float* x_in     = (const float*)d_in[0];
  const int*   nattr    = (const int*)d_in[2];
  const int*   esrc     = (const int*)d_in[3];
  const int*   edst     = (const int*)d_in[4];
  const float* emb_tab  = (const float*)d_in[5];
  const float* upM      = (const float*)d_in[6];
  const float* hArr     = (const float*)d_in[7];
  const float* mixArr   = (const float*)d_in[8];
  const float* rad_W1   = (const float*)d_in[9];
  const float* rad_b1   = (const float*)d_in[10];
  const float* rad_W2   = (const float*)d_in[11];
  const float* lin1_s   = (const float*)d_in[12];
  const float* lin1_v   = (const float*)d_in[13];
  const float* lin2_s   = (const float*)d_in[14];
  const float* lin2_v   = (const float*)d_in[15];
  const float* sc_s_w   = (const float*)d_in[16];
  const float* sc_v_w   = (const float*)d_in[17];
  const float* si_s     = (const float*)d_in[18];
  const float* si_v     = (const float*)d_in[19];
  float* out = (float*)d_out;

  float* p = (float*)d_ws;
  auto carve = [&](size_t n) { float* r = p; p += n; return r; };
  float* Q     = carve(128);
  float* emb   = carve((size_t)NN * EMBD);
  float* ys_a  = carve((size_t)NN * CC);
  float* ys_b  = carve((size_t)NN * CC);
  float* yv_a  = carve((size_t)NN * 192);
  float* yv_b  = carve((size_t)NN * 192);
  float* s1b   = carve((size_t)NN * CC);
  float* v1b   = carve((size_t)NN * 192);
  float* aggs  = carve((size_t)NN * CC);
  float* aggv  = carve((size_t)NN * 192);
  float* outs  = carve((size_t)NN * 128);
  float* outv  = carve((size_t)NN * 192);
  float* y2sb  = carve((size_t)NN * CC);
  float* y2vb  = carve((size_t)NN * 192);
  float* xb    = carve((size_t)NN * 6);

  compute_Q<<<1, 64, 0, stream>>>(upM, Q);
  init_state<<<(NN * CC) / 256, 256, 0, stream>>>(x_in, nattr, emb_tab, Q, emb,
                                                  ys_a, ys_b, yv_a, yv_b);

  float* cs = ys_a; float* os_ = ys_b;   // current / old scalar state
  float* cv = yv_a; float* ov = yv_b;    // current / old vector state
  const dim3 g64(NN / 16, 4), g128(NN / 16, 8);

  for (int i = 0; i < NLAY; ++i) {
    const float* pos = (i == 0) ? x_in : xb;
    const float* l1s = lin1_s + (size_t)i * CC * CC;
    const float* l1v = lin1_v + (size_t)i * CC * CC;
    const float* l2s = lin2_s + (size_t)i * CC * 2 * CC;
    const float* l2v = lin2_v + (size_t)i * CC * CC;
    const float* scs = sc_s_w + (size_t)i * CC * EMBD * 2 * CC;
    const float* scv = sc_v_w + (size_t)i * CC * EMBD * CC;
    const float* sis = si_s + (size_t)i * CC * CC;
    const float* siv = si_v + (size_t)i * CC * CC;
    const float* rW1 = rad_W1 + (size_t)i * NBAS * RHID;
    const float* rb1 = rad_b1 + (size_t)i * RHID;
    const float* rW2 = rad_W2 + (size_t)i * RHID * 2 * CC;

    // s1 = y_s @ lin1_s ; v1[:,:,x] = y_v[:,:,x] @ lin1_v
    wmma_gemm<<<g64, 32, 0, stream>>>(cs, CC, 1, l1s, CC, s1b, CC, 1, CC, 0);
    for (int xd = 0; xd < 3; ++xd)
      wmma_gemm<<<g64, 32, 0, stream>>>(cv + xd, 192, 3, l1v, CC, v1b + xd, 192, 3, CC, 0);

    hipMemsetAsync(aggs, 0, (size_t)NN * CC * sizeof(float), stream);
    hipMemsetAsync(aggv, 0, (size_t)NN * 192 * sizeof(float), stream);

    edge_message<<<EE / 8, 256, 0, stream>>>(pos, esrc, edst, rW1, rb1, rW2,
                                             s1b, v1b, aggs, aggv);

    // out_s = agg_s @ lin2_s + (y_s x emb) @ sc_s_w
    wmma_gemm<<<g128, 32, 0, stream>>>(aggs, CC, 1, l2s, 2 * CC, outs, 2 * CC, 1, CC, 0);
    wmma_sc_gemm<<<g128, 32, 0, stream>>>(cs, CC, 1, emb, scs, 2 * CC, outs, 2 * CC, 1);

    // out_v[:,:,x] = agg_v_x @ lin2_v + (y_v_x x emb) @ sc_v_w
    for (int xd = 0; xd < 3; ++xd) {
      wmma_gemm<<<g64, 32, 0, stream>>>(aggv + xd, 192, 3, l2v, CC, outv + xd, 192, 3, CC, 0);
      wmma_sc_gemm<<<g64, 32, 0, stream>>>(cv + xd, 192, 3, emb, scv, CC, outv + xd, 192, 3);
    }

    // self-interaction: y2_s = y_s @ si_s ; y2_v = y_v @ si_v
    wmma_gemm<<<g64, 32, 0, stream>>>(cs, CC, 1, sis, CC, y2sb, CC, 1, CC, 0);
    for (int xd = 0; xd < 3; ++xd)
      wmma_gemm<<<g64, 32, 0, stream>>>(cv + xd, 192, 3, siv, CC, y2vb + xd, 192, 3, CC, 0);

    finalize<<<(NN * CC) / 256, 256, 0, stream>>>(outs, outv, y2sb, y2vb,
                                                  cs, cv, os_, ov, hArr, mixArr, i);

    // rotate buffers: new state (just written into old buffers) becomes current
    float* t;
    t = cs; cs = os_; os_ = t;
    t = cv; cv = ov; ov = t;

    project_x<<<(NN * 6 + 255) / 256, 256, 0, stream>>>(cv, Q,
                                                        (i == NLAY - 1) ? out : xb);
  }
}